// W2LossConst_69733089018355
// MI455X (gfx1250) — compile-verified
//
#include <hip/hip_runtime.h>

// ---------------------------------------------------------------------------
// W2 loss (Wasserstein-2 misfit) for MI455X / gfx1250, wave32.
//
// Design: one 256-thread workgroup (8 waves) per trace row. Bandwidth-bound
// (64 MB total input @ 23.3 TB/s ~ 2.75us). Each element of f/obs is loaded
// exactly once via b128 loads and kept in registers. The two 2048-element
// prefix sums per row use a hierarchical scan whose 256-partial block stage
// is computed on the matrix core with V_WMMA_F32_16X16X4_F32 (full fp32, so
// no precision loss):  S = P*U + Ls*(P*J), 12 WMMAs per scan, wave 0 only
// (EXEC all-ones as WMMA requires).
// ---------------------------------------------------------------------------

typedef float v2f __attribute__((ext_vector_type(2)));
typedef float v8f __attribute__((ext_vector_type(8)));

#define NT   2048
#define NTR  4096
#define NQ   257
#define TPB  256
#define EPT  8   // elements per thread: 256*8 = 2048

__device__ __forceinline__ void wmma4(const v2f a, const v2f b, v8f& c) {
  // D = A(16x4) * B(4x16) + C, fp32. 8-arg form per CDNA5 bridge doc.
  c = __builtin_amdgcn_wmma_f32_16x16x4_f32(false, a, false, b, (short)0, c,
                                            false, false);
}

// Inclusive block scan of 256 per-thread partials using WMMA on wave 0.
// P[m][n] = partial of thread m*16+n (row-major). Inclusive scan:
//   S = P*U + Ls*(P*J)   (U upper-tri-incl ones, Ls strictly-lower ones,
//                         J all-ones -> rowsum broadcast)
// Each 16x16x16 product = 4 chained 16x16x4 WMMAs.
// Operand layout (ISA 7.12.2): A 16x4 fp32: lane l holds row m=l&15,
// regs = K {kb, kb+1}, kb = 2*(l>>4). C/D: reg r = row r+8*(l>>4), col l&15.
// B 4x16 assumed mirror of A: reg j = row kb+j, col l&15.
__device__ __forceinline__ void block_scan(float local_sum,
                                           float* s_part, float* s_incl,
                                           float* s_rs,
                                           float& excl, float& total) {
  const int tid = threadIdx.x;
  s_part[tid] = local_sum;
  __syncthreads();

  const int mrow = tid & 15;
  const int half = (tid >> 4) & 1;
  const int kb   = half << 1;       // 0 or 2
  const int ncol = tid & 15;

  v2f ap[4];
  v8f acc = {0.f, 0.f, 0.f, 0.f, 0.f, 0.f, 0.f, 0.f};

  if (tid < 32) {   // wave 0, EXEC all 1s
    #pragma unroll
    for (int c = 0; c < 4; ++c) {
      ap[c].x = s_part[mrow * 16 + 4 * c + kb];
      ap[c].y = s_part[mrow * 16 + 4 * c + kb + 1];
    }
    // T2 = P * J : every column = rowsum[m]
    v8f t2 = {0.f, 0.f, 0.f, 0.f, 0.f, 0.f, 0.f, 0.f};
    v2f bone; bone.x = 1.0f; bone.y = 1.0f;
    #pragma unroll
    for (int c = 0; c < 4; ++c) wmma4(ap[c], bone, t2);
    if (ncol == 0) {  // lanes 0 and 16 own column 0 of rows 0-7 / 8-15
      #pragma unroll
      for (int r = 0; r < 8; ++r) s_rs[r + 8 * half] = t2[r];
    }
    // acc = P * U  (U[k][n] = k <= n)
    #pragma unroll
    for (int c = 0; c < 4; ++c) {
      v2f bu;
      bu.x = (4 * c + kb     <= ncol) ? 1.0f : 0.0f;
      bu.y = (4 * c + kb + 1 <= ncol) ? 1.0f : 0.0f;
      wmma4(ap[c], bu, acc);
    }
  }
  __syncthreads();   // publish s_rs to all lanes of wave 0
  if (tid < 32) {
    // acc += Ls * T2  (Ls[m][k] = k < m ; T2[k][n] = rowsum[k])
    #pragma unroll
    for (int c = 0; c < 4; ++c) {
      v2f al, bt;
      al.x = (4 * c + kb     < mrow) ? 1.0f : 0.0f;
      al.y = (4 * c + kb + 1 < mrow) ? 1.0f : 0.0f;
      bt.x = s_rs[4 * c + kb];
      bt.y = s_rs[4 * c + kb + 1];
      wmma4(al, bt, acc);
    }
    #pragma unroll
    for (int r = 0; r < 8; ++r) s_incl[(r + 8 * half) * 16 + ncol] = acc[r];
  }
  __syncthreads();
  const float incl = s_incl[tid];
  excl  = incl - local_sum;
  total = s_incl[TPB - 1];
}

__global__ void __launch_bounds__(TPB)
w2_row_kernel(const float* __restrict__ f, const float* __restrict__ obs,
              const float* __restrict__ t, const float* __restrict__ p,
              float* __restrict__ row_out) {
  __shared__ float s_part[TPB];
  __shared__ float s_incl[TPB];
  __shared__ float s_rs[16];
  __shared__ float s_cdf[NT];
  __shared__ float s_p[NQ];
  __shared__ float s_q[NQ];
  __shared__ float s_h[NQ - 1];
  __shared__ float s_slope[NQ - 1];
  __shared__ float s_cp[NQ - 2];
  __shared__ float s_dp[NQ - 2];
  __shared__ float s_M[NQ];
  __shared__ float s_a[NQ - 1], s_b[NQ - 1], s_c[NQ - 1], s_d[NQ - 1];
  __shared__ float s_scal[8];
  __shared__ float s_red[TPB];

  const int row = blockIdx.x;
  const int tid = threadIdx.x;
  const float dx = t[1] - t[0];

  for (int j = tid; j < NQ; j += TPB) s_p[j] = p[j];  // covered by scan barrier

  // ---------------- Phase A: obs -> cdf -> quantiles q ----------------
  const float4* obs4 = reinterpret_cast<const float4*>(obs + (size_t)row * NT);
  float gv[EPT];
  {
    float4 a0 = obs4[tid * 2], a1 = obs4[tid * 2 + 1];
    gv[0] = a0.x * a0.x; gv[1] = a0.y * a0.y; gv[2] = a0.z * a0.z; gv[3] = a0.w * a0.w;
    gv[4] = a1.x * a1.x; gv[5] = a1.y * a1.y; gv[6] = a1.z * a1.z; gv[7] = a1.w * a1.w;
  }
  float lp[EPT], run = 0.f;
  #pragma unroll
  for (int j = 0; j < EPT; ++j) { run += gv[j]; lp[j] = run; }

  float excl, total;
  block_scan(run, s_part, s_incl, s_rs, excl, total);

  if (tid == 0)       s_scal[0] = gv[0];
  if (tid == TPB - 1) s_scal[1] = gv[EPT - 1];
  __syncthreads();
  const float g0 = s_scal[0], gl = s_scal[1];
  const float invTg = 1.0f / ((total - 0.5f * (g0 + gl)) * dx);

  // cumtrapz[i] = 0.5*dx*(2*S_i - g_i - g_0); cdf = clip(cumtrapz/Tg, 0, 1)
  #pragma unroll
  for (int j = 0; j < EPT; ++j) {
    float S = excl + lp[j];
    float c = 0.5f * dx * (2.0f * S - gv[j] - g0) * invTg;
    s_cdf[tid * EPT + j] = fminf(fmaxf(c, 0.0f), 1.0f);
  }
  __syncthreads();

  // searchsorted(cdf, p_j, 'left'), q = t[clip(idx, 0, NT-1)]
  for (int j = tid; j < NQ; j += TPB) {
    const float target = s_p[j];
    int lo = 0, hi = NT;
    while (lo < hi) {
      int mid = (lo + hi) >> 1;
      if (s_cdf[mid] < target) lo = mid + 1; else hi = mid;
    }
    if (lo > NT - 1) lo = NT - 1;
    s_q[j] = t[lo];
  }
  __syncthreads();

  // ---------------- Natural cubic spline coefficients ----------------
  for (int j = tid; j < NQ - 1; j += TPB) {
    const float h = s_p[j + 1] - s_p[j];
    s_h[j]     = h;
    s_slope[j] = (s_q[j + 1] - s_q[j]) / h;
  }
  __syncthreads();

  if (tid == 0) {
    // Thomas algorithm on the 255x255 tridiagonal system. Row i:
    // sub = h[i], diag = 2*(h[i]+h[i+1]), sup = h[i+1], rhs = 6*(slope[i+1]-slope[i])
    const int n = NQ - 2;  // 255
    float diag0 = 2.0f * (s_h[0] + s_h[1]);
    s_cp[0] = s_h[1] / diag0;
    s_dp[0] = 6.0f * (s_slope[1] - s_slope[0]) / diag0;
    for (int i = 1; i < n; ++i) {
      float sub  = s_h[i];
      float w    = 2.0f * (s_h[i] + s_h[i + 1]) - sub * s_cp[i - 1];
      float inv  = 1.0f / w;
      float rhs  = 6.0f * (s_slope[i + 1] - s_slope[i]);
      s_cp[i] = s_h[i + 1] * inv;
      s_dp[i] = (rhs - sub * s_dp[i - 1]) * inv;
    }
    float m = s_dp[n - 1];
    s_M[n] = m;                      // M[1..255] = M_inner[0..254]
    for (int i = n - 2; i >= 0; --i) { m = s_dp[i] - s_cp[i] * m; s_M[i + 1] = m; }
    s_M[0] = 0.0f; s_M[NQ - 1] = 0.0f;
  }
  __syncthreads();

  {
    const int j = tid;  // NQ-1 == TPB == 256
    const float h = s_h[j], Mj = s_M[j], Mj1 = s_M[j + 1];
    s_a[j] = s_q[j];
    s_b[j] = s_slope[j] - h * (2.0f * Mj + Mj1) * (1.0f / 6.0f);
    s_c[j] = 0.5f * Mj;
    s_d[j] = (Mj1 - Mj) / (6.0f * h);
  }
  __syncthreads();

  // ---------------- Phase B: F = cumtrapz(renorm(f)), spline eval ----------
  const float4* f4 = reinterpret_cast<const float4*>(f + (size_t)row * NT);
  float xv[EPT];
  {
    float4 a0 = f4[tid * 2], a1 = f4[tid * 2 + 1];
    xv[0] = a0.x; xv[1] = a0.y; xv[2] = a0.z; xv[3] = a0.w;
    xv[4] = a1.x; xv[5] = a1.y; xv[6] = a1.z; xv[7] = a1.w;
  }
  float g2[EPT], lp2[EPT], run2 = 0.f;
  #pragma unroll
  for (int j = 0; j < EPT; ++j) { g2[j] = xv[j] * xv[j]; run2 += g2[j]; lp2[j] = run2; }

  float excl2, total2;
  block_scan(run2, s_part, s_incl, s_rs, excl2, total2);

  if (tid == 0)       s_scal[2] = g2[0];
  if (tid == TPB - 1) s_scal[3] = g2[EPT - 1];
  __syncthreads();
  const float q0 = s_scal[2], ql = s_scal[3];
  const float invTf = 1.0f / ((total2 - 0.5f * (q0 + ql)) * dx);

  const float4* t4 = reinterpret_cast<const float4*>(t);
  float tv[EPT];
  {
    float4 a0 = t4[tid * 2], a1 = t4[tid * 2 + 1];
    tv[0] = a0.x; tv[1] = a0.y; tv[2] = a0.z; tv[3] = a0.w;
    tv[4] = a1.x; tv[5] = a1.y; tv[6] = a1.z; tv[7] = a1.w;
  }

  const float p0 = s_p[0];
  const float invh = (float)(NQ - 1) / (s_p[NQ - 1] - s_p[0]);
  float csum = 0.f, c_first = 0.f, c_last = 0.f;
  #pragma unroll
  for (int j = 0; j < EPT; ++j) {
    float S = excl2 + lp2[j];
    float F = 0.5f * dx * (2.0f * S - g2[j] - q0) * invTf;  // not clipped (ref)
    // k = searchsorted(p, F, 'left'): arithmetic guess + local fixup
    int k = (int)floorf((F - p0) * invh);
    k = min(max(k, 0), NQ - 1);
    while (k < NQ && s_p[k] < F) ++k;
    while (k > 0 && s_p[k - 1] >= F) --k;
    int idx = min(max(k - 1, 0), NQ - 2);
    float fr = F - s_p[idx];
    float sv = s_a[idx] + fr * (s_b[idx] + fr * (s_c[idx] + fr * s_d[idx]));
    float df = tv[j] - sv;
    float contrib = df * df * xv[j];
    csum += contrib;
    if (j == 0)       c_first = contrib;
    if (j == EPT - 1) c_last  = contrib;
  }
  if (tid == 0)       s_scal[4] = c_first;
  if (tid == TPB - 1) s_scal[5] = c_last;

  s_red[tid] = csum;
  __syncthreads();
  #pragma unroll
  for (int off = TPB / 2; off > 0; off >>= 1) {
    if (tid < off) s_red[tid] += s_red[tid + off];
    __syncthreads();
  }
  if (tid == 0) {
    row_out[row] = (s_red[0] - 0.5f * (s_scal[4] + s_scal[5])) * dx;
  }
}

// Deterministic final reduction of the 4096 per-row scalars.
__global__ void __launch_bounds__(TPB)
w2_reduce_kernel(const float* __restrict__ rows, float* __restrict__ out) {
  __shared__ float s[TPB];
  const int tid = threadIdx.x;
  float acc = 0.f;
  for (int i = tid; i < NTR; i += TPB) acc += rows[i];
  s[tid] = acc;
  __syncthreads();
  #pragma unroll
  for (int off = TPB / 2; off > 0; off >>= 1) {
    if (tid < off) s[tid] += s[tid + off];
    __syncthreads();
  }
  if (tid == 0) out[0] = s[0];
}

extern "C" void kernel_launch(void* const* d_in, const int* in_sizes, int n_in,
                              void* d_out, int out_size, void* d_ws, size_t ws_size,
                              hipStream_t stream) {
  const float* f   = (const float*)d_in[0];
  const float* obs = (const float*)d_in[1];
  const float* t   = (const float*)d_in[2];
  const float* p   = (const float*)d_in[3];
  float* ws = (float*)d_ws;  // 4096 floats of per-row results

  w2_row_kernel<<<NTR, TPB, 0, stream>>>(f, obs, t, p, ws);
  w2_reduce_kernel<<<1, TPB, 0, stream>>>(ws, (float*)d_out);
}